// Encoder_1537598292364
// MI455X (gfx1250) — compile-verified
//
#include <hip/hip_runtime.h>
#include <math.h>

// Masked LSTM encoder for MI455X (gfx1250, wave32).
// B=64, T=1024, V=50000, E=256, H=512, 4H=2048.
//
// Strategy:
//  - Persistent kernel: 64 workgroups x 256 threads (8 waves). WG g owns
//    hidden units [8g, 8g+8) -> 32 gate columns (i|f in N-tile 0, g|o in
//    N-tile 1). Per timestep each wave computes one 16x16 fp32 WMMA tile
//    over K=768 ( [e_t | h_{t-1}] @ [kernel ; rec_kernel] ), using
//    V_WMMA_F32_16X16X4_F32 for full fp32 precision (cell state integrates
//    1024 steps; bf16/f16 inputs would drift).
//  - Weights are pre-swizzled once into the exact per-lane B-fragment layout
//    so every B fragment is one coalesced global_load_b64 (L2-resident, 6MB).
//  - Embedding gather is fused into the A-fragment load (no 512MB xz buffer).
//  - h is double-buffered in workspace; a device-wide atomic barrier
//    separates timesteps (scan is sync-bound, not BW-bound: total HBM
//    traffic ~136MB vs 23.3 TB/s).

#define B_   64
#define T_   1024
#define E_   256
#define H_   512
#define G4_  2048
#define NWG  64
#define KTOT 768
#define NKT  (KTOT/4)   // 192 K-tiles of depth 4
#define NKTE (E_/4)     // 64 K-tiles in embedding part

typedef __attribute__((ext_vector_type(2))) float v2f;
typedef __attribute__((ext_vector_type(8))) float v8f;

__device__ __forceinline__ float sigf(float v) {
    return 1.0f / (1.0f + __expf(-v));
}

// ---- device-wide split barrier (persistent-kernel sync) ------------------
__device__ __forceinline__ void grid_barrier(unsigned* bar) {
    __syncthreads();
    if (threadIdx.x == 0) {
        unsigned phase = __hip_atomic_load(&bar[1], __ATOMIC_ACQUIRE,
                                           __HIP_MEMORY_SCOPE_AGENT);
        unsigned arrived = __hip_atomic_fetch_add(&bar[0], 1u, __ATOMIC_ACQ_REL,
                                                  __HIP_MEMORY_SCOPE_AGENT);
        if (arrived == NWG - 1) {
            __hip_atomic_store(&bar[0], 0u, __ATOMIC_RELAXED,
                               __HIP_MEMORY_SCOPE_AGENT);
            __hip_atomic_store(&bar[1], phase + 1u, __ATOMIC_RELEASE,
                               __HIP_MEMORY_SCOPE_AGENT);
        } else {
            while (__hip_atomic_load(&bar[1], __ATOMIC_ACQUIRE,
                                     __HIP_MEMORY_SCOPE_AGENT) == phase) {
                __builtin_amdgcn_s_sleep(2);
            }
        }
    }
    __syncthreads();
}

// ---- one-time weight swizzle into per-lane WMMA B-fragment layout --------
// packed[((wg*2+nt)*NKT + kt)*32 + lane] holds 2 floats:
//   lane<16 : (K=kt*4+0, N=lane), (K=kt*4+1, N=lane)
//   lane>=16: (K=kt*4+2, N=lane-16), (K=kt*4+3, N=lane-16)
// N-column mapping: nt=0 -> [i(0..7) | f(8..15)], nt=1 -> [g | o] of the
// 8-wide hidden slice owned by workgroup wg.
__global__ void lstm_pack(const float* __restrict__ kern,
                          const float* __restrict__ rec,
                          float* __restrict__ packed,
                          unsigned* __restrict__ bar) {
    if (blockIdx.x == 0 && threadIdx.x == 0) { bar[0] = 0u; bar[1] = 0u; }
    int tid  = blockIdx.x * blockDim.x + threadIdx.x;   // [0, NWG*2*NKT*32)
    int lane = tid & 31;
    int r    = tid >> 5;
    int kt   = r % NKT;
    int r2   = r / NKT;
    int nt   = r2 & 1;
    int wg   = r2 >> 1;
    if (wg >= NWG) return;

    int ncl  = lane & 15;                 // column within 16-wide tile
    int gate = nt * 2 + (ncl >> 3);       // 0=i,1=f,2=g,3=o
    int gcol = gate * H_ + wg * 8 + (ncl & 7);
    int k0   = kt * 4 + ((lane >> 4) << 1);

    float w0, w1;
    if (k0 < E_) {
        w0 = kern[(size_t)k0 * G4_ + gcol];
        w1 = kern[(size_t)(k0 + 1) * G4_ + gcol];
    } else {
        int kk = k0 - E_;
        w0 = rec[(size_t)kk * G4_ + gcol];
        w1 = rec[(size_t)(kk + 1) * G4_ + gcol];
    }
    packed[(size_t)tid * 2]     = w0;
    packed[(size_t)tid * 2 + 1] = w1;
}

// ---- persistent masked-LSTM scan -----------------------------------------
__global__ __launch_bounds__(256)
void lstm_scan(const int*   __restrict__ x,
               const float* __restrict__ h0,
               const float* __restrict__ c0,
               const float* __restrict__ emb,
               const float* __restrict__ bias,
               const float* __restrict__ packed,
               float* __restrict__ ys,
               float* __restrict__ hT,
               float* __restrict__ cT,
               float* __restrict__ hbuf,   // 2 * B_*H_ double buffer
               unsigned* __restrict__ bar) {
    __shared__ float zbuf[B_][33];   // 64x32 gate pre-activations (+pad)
    __shared__ float c_s[B_][8];     // this WG's cell-state slice

    const int wg   = blockIdx.x;
    const int tid  = threadIdx.x;
    const int lane = tid & 31;
    const int w    = tid >> 5;          // wave 0..7
    const int m0   = (w & 3) << 4;      // batch-row tile base
    const int nt   = w >> 2;            // N-tile 0/1
    const int hu0  = wg * 8;            // hidden-unit slice base

    // init cell-state slice from c0
    for (int i = tid; i < B_ * 8; i += 256) {
        int b = i >> 3, hl = i & 7;
        c_s[b][hl] = c0[b * H_ + hu0 + hl];
    }

    // hoist per-thread bias values for the elementwise stage
    float bi[2], bf2[2], bg[2], bo[2];
    #pragma unroll
    for (int j = 0; j < 2; ++j) {
        int hl = (tid + j * 256) & 7;
        bi[j]  = bias[hu0 + hl];
        bf2[j] = bias[H_ + hu0 + hl];
        bg[j]  = bias[2 * H_ + hu0 + hl];
        bo[j]  = bias[3 * H_ + hu0 + hl];
    }

    const int arow = m0 + (lane & 15);        // A-fragment batch row
    const int koff = (lane >> 4) << 1;        // 0 or 2 within K-tile
    const float* wbase = packed + (size_t)(wg * 2 + nt) * NKT * 64;

    __syncthreads();

    for (int t = 0; t < T_; ++t) {
        const float* hsrc = (t == 0) ? h0
                                     : (hbuf + (size_t)((t & 1) ^ 1) * (B_ * H_));
        float* hdst = hbuf + (size_t)(t & 1) * (B_ * H_);

        // A-fragment source pointers: fused embedding gather + recurrent h
        const int tok = x[arow * T_ + t];
        const float* ea = emb + (size_t)tok * E_ + koff;
        const float* ha = hsrc + (size_t)arow * H_ + koff;
        const float* wp = wbase + lane * 2;

        v8f acc = {};
        #pragma unroll 4
        for (int kt = 0; kt < NKTE; ++kt) {          // K = embedding part
            v2f a  = *(const v2f*)(ea + kt * 4);
            v2f bm = *(const v2f*)(wp); wp += 64;
            acc = __builtin_amdgcn_wmma_f32_16x16x4_f32(
                false, a, false, bm, (short)0, acc, false, false);
        }
        #pragma unroll 4
        for (int kt = 0; kt < NKT - NKTE; ++kt) {    // K = recurrent part
            v2f a  = *(const v2f*)(ha + kt * 4);
            v2f bm = *(const v2f*)(wp); wp += 64;
            acc = __builtin_amdgcn_wmma_f32_16x16x4_f32(
                false, a, false, bm, (short)0, acc, false, false);
        }

        // spill 16x16 C tile to LDS (C layout: VGPR r, lane<16 -> M=r,
        // lane>=16 -> M=8+r, N=lane%16)
        {
            int row = m0 + ((lane >> 4) << 3);
            int col = (nt << 4) + (lane & 15);
            #pragma unroll
            for (int r = 0; r < 8; ++r) zbuf[row + r][col] = acc[r];
        }
        __syncthreads();

        // gates + masked state update (2 (b,hu) pairs per thread)
        #pragma unroll
        for (int j = 0; j < 2; ++j) {
            int i  = tid + j * 256;
            int b  = i >> 3, hl = i & 7;
            float zi = zbuf[b][hl]      + bi[j];
            float zf = zbuf[b][8 + hl]  + bf2[j];
            float zg = zbuf[b][16 + hl] + bg[j];
            float zo = zbuf[b][24 + hl] + bo[j];
            float ig = sigf(zi), fg = sigf(zf), og = sigf(zo);
            float gg = tanhf(zg);
            float cold = c_s[b][hl];
            float cnew = fg * cold + ig * gg;
            float hnew = og * tanhf(cnew);
            bool  mt   = (x[b * T_ + t] != 0);
            float hold = hsrc[b * H_ + hu0 + hl];
            float hk   = mt ? hnew : hold;
            float ck   = mt ? cnew : cold;
            c_s[b][hl] = ck;
            hdst[b * H_ + hu0 + hl] = hk;
            ys[((size_t)b * T_ + t) * H_ + hu0 + hl] = hk;
            if (t == T_ - 1) {
                hT[b * H_ + hu0 + hl] = hk;
                cT[b * H_ + hu0 + hl] = ck;
            }
        }

        grid_barrier(bar);   // all WGs see new h before next step
    }
}

extern "C" void kernel_launch(void* const* d_in, const int* in_sizes, int n_in,
                              void* d_out, int out_size, void* d_ws, size_t ws_size,
                              hipStream_t stream) {
    const int*   x    = (const int*)d_in[0];
    const float* h0   = (const float*)d_in[1];
    const float* c0   = (const float*)d_in[2];
    const float* emb  = (const float*)d_in[3];
    const float* kern = (const float*)d_in[4];
    const float* rec  = (const float*)d_in[5];
    const float* bias = (const float*)d_in[6];

    float* ys = (float*)d_out;
    float* hT = ys + (size_t)B_ * T_ * H_;
    float* cT = hT + (size_t)B_ * H_;

    unsigned char* ws = (unsigned char*)d_ws;
    unsigned* bar  = (unsigned*)ws;                       // 256 B
    float* packed  = (float*)(ws + 256);                  // 6 MB swizzled W
    float* hbuf    = packed + (size_t)NWG * 2 * NKT * 64; // 256 KB h dbl-buf

    const int pack_threads = NWG * 2 * NKT * 32;          // 786432
    lstm_pack<<<pack_threads / 256, 256, 0, stream>>>(kern, rec, packed, bar);
    lstm_scan<<<NWG, 256, 0, stream>>>(x, h0, c0, emb, bias, packed,
                                       ys, hT, cT, hbuf, bar);
}